// ClassifierHead_78752520339712
// MI455X (gfx1250) — compile-verified
//
#include <hip/hip_runtime.h>

typedef float v2f __attribute__((ext_vector_type(2)));
typedef float v8f __attribute__((ext_vector_type(8)));

#define BB    64
#define CIN   512
#define TT    1024
#define COUT  256

#define O_TILE  64
#define T_CHUNK 32
#define WROW    516   // padded LDS row stride (floats) for W and X tiles
#define YROW    33    // padded LDS row stride for Y tile

#define XOFF (O_TILE * WROW)                 // 33024
#define YOFF (XOFF + T_CHUNK * WROW)         // 49536
#define SMEM_FLOATS (YOFF + O_TILE * YROW)   // 51648 floats = 206592 bytes

__global__ __launch_bounds__(256)
void lif_head_fused(const float* __restrict__ x, const float* __restrict__ W,
                    const float* __restrict__ gamma, const float* __restrict__ beta,
                    const float* __restrict__ rmean, const float* __restrict__ rvar,
                    float* __restrict__ out)
{
    extern __shared__ float smem[];
    float* sW = smem;          // [64][WROW]  W tile, row = o, col = c
    float* sX = smem + XOFF;   // [32][WROW]  x chunk, row = t, col = c (transposed stage)
    float* sY = smem + YOFF;   // [64][YROW]  y chunk, row = o, col = t

    const int tid  = threadIdx.x;
    const int wave = tid >> 5;
    const int lane = tid & 31;
    const int b    = blockIdx.x >> 2;
    const int o0   = (blockIdx.x & 3) * O_TILE;

    // ---- stage W o-tile into LDS once (reused for all 32 t-chunks) ----
    for (int i = tid; i < O_TILE * (CIN / 4); i += 256) {
        const int o  = i >> 7;            // /128 float4 per row
        const int c4 = (i & 127) << 2;
        const float4 w4 = *(const float4*)(W + (size_t)(o0 + o) * CIN + c4);
        *(float4*)(sW + o * WROW + c4) = w4;   // row start 2064B-aligned, c4 16B-aligned
    }

    // ---- per-o folded BN params + LIF state (scan threads: tid < 64) ----
    float scale = 0.f, bias = 0.f, v = 0.f, cnt = 0.f;
    if (tid < O_TILE) {
        const int o = o0 + tid;
        scale = gamma[o] * rsqrtf(rvar[o] + 1e-5f);
        bias  = beta[o] - rmean[o] * scale;
    }
    __syncthreads();

    const int m_tile = wave & 3;           // which o sub-tile of 16
    const int n_tile = wave >> 2;          // which t sub-tile of 16
    const int l15    = lane & 15;
    const int khalf  = (lane >> 4) << 1;   // lanes 0-15 -> K {0,1}; lanes 16-31 -> K {2,3}

    const float* aBase = sW + (m_tile * 16 + l15) * WROW + khalf;   // A: 16x4 (o x c)
    const float* bBase = sX + (n_tile * 16 + l15) * WROW + khalf;   // B: 4x16 (c x t)

    for (int chunk = 0; chunk < TT / T_CHUNK; ++chunk) {
        const int t0 = chunk * T_CHUNK;

        // ---- stage x chunk transposed into LDS: sX[t][c] ----
        {
            const int tseg = (tid & 7) << 2;         // 4 t-values per thread
            const int crow = tid >> 3;               // 32 c-rows per pass
            for (int p = 0; p < CIN / 32; ++p) {
                const int c = p * 32 + crow;
                const float4 xv = *(const float4*)(x + ((size_t)(b * CIN + c)) * TT + t0 + tseg);
                sX[(tseg + 0) * WROW + c] = xv.x;
                sX[(tseg + 1) * WROW + c] = xv.y;
                sX[(tseg + 2) * WROW + c] = xv.z;
                sX[(tseg + 3) * WROW + c] = xv.w;
            }
        }
        __syncthreads();

        // ---- fp32 WMMA GEMM: D[16 o x 16 t] += A[16x4] * B[4x16] over Cin ----
        // Two independent accumulators (even/odd K-quads) break the serial
        // D->C dependency chain between consecutive v_wmma issues.
        v8f acc0 = {};
        v8f acc1 = {};
        #pragma unroll 2
        for (int kk = 0; kk < CIN; kk += 8) {
            const v2f a0 = *(const v2f*)(aBase + kk);       // ds_load_b64, 8B aligned
            const v2f b0 = *(const v2f*)(bBase + kk);
            const v2f a1 = *(const v2f*)(aBase + kk + 4);
            const v2f b1 = *(const v2f*)(bBase + kk + 4);
            acc0 = __builtin_amdgcn_wmma_f32_16x16x4_f32(
                       false, a0, false, b0, (short)0, acc0, false, false);
            acc1 = __builtin_amdgcn_wmma_f32_16x16x4_f32(
                       false, a1, false, b1, (short)0, acc1, false, false);
        }

        // ---- spill D tile to sY: VGPR r -> M=r (lanes 0-15) / M=r+8 (lanes 16-31) ----
        {
            const int orow = m_tile * 16 + ((lane >> 4) << 3);
            const int tcol = n_tile * 16 + l15;
            #pragma unroll
            for (int r = 0; r < 8; ++r)
                sY[(orow + r) * YROW + tcol] = acc0[r] + acc1[r];
        }
        __syncthreads();

        // ---- LIF scan over this chunk: one thread per output channel ----
        if (tid < O_TILE) {
            const float* yrow = sY + tid * YROW;
            #pragma unroll 8
            for (int t = 0; t < T_CHUNK; ++t) {
                const float y = yrow[t] * scale + bias;  // folded BatchNorm
                v = v + (y - v) * 0.5f;                  // charge, TAU = 2
                const float s = (v >= 1.0f) ? 1.0f : 0.0f;
                cnt += s;
                v *= (1.0f - s);                         // hard reset
            }
        }
        __syncthreads();   // protect sX/sY before next chunk restages
    }

    if (tid < O_TILE)
        out[(size_t)b * COUT + o0 + tid] = cnt * (1.0f / (float)TT);
}

extern "C" void kernel_launch(void* const* d_in, const int* in_sizes, int n_in,
                              void* d_out, int out_size, void* d_ws, size_t ws_size,
                              hipStream_t stream) {
    (void)in_sizes; (void)n_in; (void)out_size; (void)d_ws; (void)ws_size;
    const float* x     = (const float*)d_in[0];
    const float* W     = (const float*)d_in[1];
    const float* gamma = (const float*)d_in[2];
    const float* beta  = (const float*)d_in[3];
    const float* rmean = (const float*)d_in[4];
    const float* rvar  = (const float*)d_in[5];
    float* out = (float*)d_out;

    const size_t shmem = SMEM_FLOATS * sizeof(float);   // ~202 KB, within 320 KB/WGP
    static bool attr_set = false;  // idempotent host-side attribute, not a stream op
    if (!attr_set) {
        hipFuncSetAttribute((const void*)lif_head_fused,
                            hipFuncAttributeMaxDynamicSharedMemorySize, (int)shmem);
        attr_set = true;
    }

    // 256 workgroups: (b, o-tile) pairs; each walks T sequentially (LIF carry).
    lif_head_fused<<<dim3(BB * (COUT / O_TILE)), dim3(256), shmem, stream>>>(
        x, W, gamma, beta, rmean, rvar, out);
}